// Knn_36077725286459
// MI455X (gfx1250) — compile-verified
//
#include <hip/hip_runtime.h>

// CDNA5 / gfx1250, wave32. fp32 WMMA path: V_WMMA_F32_16X16X4_F32.
typedef __attribute__((ext_vector_type(2))) float v2f;
typedef __attribute__((ext_vector_type(8))) float v8f;

#define BATCH 16
#define CH    64
#define NPTS  3136           // 56*56
#define KNN   16
#define TILE  16
#define NT    (NPTS / TILE)  // 196 column/row tiles
#define LDS_STRIDE (NPTS + 4)  // pad: 8*3140 % 64 == 32 -> row halves hit disjoint banks

// ---------------------------------------------------------------------------
// Phase 1: L2-normalize each point along C; emit col-major [B][C][N] (B-operand),
// row-major [B][N][C] (A-operand), and per-point sum(xn^2).
// ---------------------------------------------------------------------------
__global__ void __launch_bounds__(256)
knn_normalize(const float* __restrict__ x,
              float* __restrict__ pn_col,
              float* __restrict__ pn_row,
              float* __restrict__ sq)
{
    int p = blockIdx.x * blockDim.x + threadIdx.x;   // 0 .. B*N-1
    int b = p / NPTS;
    int n = p - b * NPTS;
    const float* xb = x + (size_t)b * CH * NPTS + n;

    float v[CH];
    float ss = 0.f;
    #pragma unroll
    for (int c = 0; c < CH; ++c) {
        v[c] = xb[(size_t)c * NPTS];
        ss += v[c] * v[c];
    }
    float scale = 1.0f / fmaxf(sqrtf(ss), 1e-12f);
    float s2 = 0.f;
    #pragma unroll
    for (int c = 0; c < CH; ++c) {
        v[c] *= scale;
        s2 += v[c] * v[c];
    }
    sq[p] = s2;

    float* pc = pn_col + (size_t)b * CH * NPTS + n;
    #pragma unroll
    for (int c = 0; c < CH; ++c) pc[(size_t)c * NPTS] = v[c];

    float* pr = pn_row + (size_t)p * CH;             // 256B-aligned per point
    #pragma unroll
    for (int c = 0; c < CH; c += 4)
        *(float4*)(pr + c) = make_float4(v[c], v[c+1], v[c+2], v[c+3]);
}

// ---------------------------------------------------------------------------
// Phase 2: one workgroup (16 waves) per (batch, 16-row stripe).
//   A) waves cooperatively fill the 16 x 3136 distance stripe in LDS via
//      v_wmma_f32_16x16x4_f32 Gram tiles.
//   B) wave w extracts top-16 (smallest, lowest-index tie-break) of row w.
// ---------------------------------------------------------------------------
__global__ void __launch_bounds__(512)
knn_tile(const float* __restrict__ pn_col,
         const float* __restrict__ pn_row,
         const float* __restrict__ sq,
         int* __restrict__ out)
{
    extern __shared__ float dist[];                  // [16][LDS_STRIDE]
    int bb   = blockIdx.x / NT;
    int rt   = blockIdx.x - bb * NT;
    int row0 = rt * TILE;

    int wave = threadIdx.x >> 5;
    int lane = threadIdx.x & 31;
    int half = lane >> 4;                            // K-pair select (A/B layout)
    int l16  = lane & 15;                            // M (A) / N (B,C/D) index

    const float* prow = pn_row + ((size_t)bb * NPTS + row0) * CH;
    const float* pcol = pn_col + (size_t)bb * CH * NPTS;
    const float* sqb  = sq + (size_t)bb * NPTS;

    // Preload the 16 A-fragments for this row stripe (K chunks of 4).
    // ISA 16x4 f32 A layout: lanes 0-15 hold K={0,1}, lanes 16-31 K={2,3}.
    v2f afrag[16];
    #pragma unroll
    for (int kc = 0; kc < 16; ++kc)
        afrag[kc] = *(const v2f*)(prow + (size_t)l16 * CH + kc * 4 + half * 2);

    // --- Phase A: Gram tiles -> distances -> LDS ---
    for (int t = wave; t < NT; t += 16) {
        int c0 = t * TILE;
        v8f acc = {};
        #pragma unroll
        for (int kc = 0; kc < 16; ++kc) {
            int k0 = kc * 4 + half * 2;              // B 4x16 layout mirrors A
            v2f bfrag;
            bfrag.x = pcol[(size_t)k0       * NPTS + c0 + l16];
            bfrag.y = pcol[(size_t)(k0 + 1) * NPTS + c0 + l16];
            acc = __builtin_amdgcn_wmma_f32_16x16x4_f32(
                false, afrag[kc], false, bfrag, (short)0, acc, false, false);
        }
        float sqcol = sqb[c0 + l16];
        // C/D layout: VGPR r -> row (r + 8*half), lane l16 -> column c0+l16.
        #pragma unroll
        for (int r = 0; r < 8; ++r) {
            int m = r + half * 8;
            float d = sqb[row0 + m] + sqcol - 2.0f * acc[r];
            dist[m * LDS_STRIDE + c0 + l16] = d;
        }
    }
    __syncthreads();

    // --- Phase B: top-16 per row (wave w owns row w) ---
    float* drow = dist + wave * LDS_STRIDE;
    int rowg = row0 + wave;
    int* out0 = out + ((size_t)bb * NPTS + rowg) * KNN;                 // nn_idx
    int* out1 = out + ((size_t)(BATCH + bb) * NPTS + rowg) * KNN;       // center

    for (int j = 0; j < KNN; ++j) {
        float best = 3.0e38f;
        int   bidx = NPTS;
        for (int c = lane; c < NPTS; c += 32) {      // 98 elems/lane, idx ascending
            float v = drow[c];
            if (v < best) { best = v; bidx = c; }
        }
        #pragma unroll
        for (int off = 16; off > 0; off >>= 1) {     // butterfly -> all lanes agree
            float ov = __shfl_xor(best, off, 32);
            int   oi = __shfl_xor(bidx, off, 32);
            if (ov < best || (ov == best && oi < bidx)) { best = ov; bidx = oi; }
        }
        if (lane == 0) { out0[j] = bidx; out1[j] = rowg; }
        if (lane == (bidx & 31)) drow[bidx] = 3.0e38f;   // remove winner
    }
}

// ---------------------------------------------------------------------------
extern "C" void kernel_launch(void* const* d_in, const int* in_sizes, int n_in,
                              void* d_out, int out_size, void* d_ws, size_t ws_size,
                              hipStream_t stream)
{
    (void)in_sizes; (void)n_in; (void)out_size; (void)ws_size;
    const float* x = (const float*)d_in[0];          // d_in[1] = k (==16, baked in)

    float* pn_col = (float*)d_ws;                                  // 12.85 MB
    float* pn_row = pn_col + (size_t)BATCH * CH * NPTS;            // 12.85 MB
    float* sqv    = pn_row + (size_t)BATCH * CH * NPTS;            // 0.20 MB
    int*   out    = (int*)d_out;                                   // [2][B][N][K] int32

    knn_normalize<<<(BATCH * NPTS) / 256, 256, 0, stream>>>(x, pn_col, pn_row, sqv);

    size_t lds_bytes = (size_t)TILE * LDS_STRIDE * sizeof(float);  // 200,960 B < 320 KB
    knn_tile<<<BATCH * NT, 512, lds_bytes, stream>>>(pn_col, pn_row, sqv, out);
}